// Foveation_31568009625658
// MI455X (gfx1250) — compile-verified
//
#include <hip/hip_runtime.h>
#include <hip/hip_bf16.h>

typedef float v2f __attribute__((ext_vector_type(2)));
typedef float v8f __attribute__((ext_vector_type(8)));

// ---------------- helpers ----------------

__device__ __forceinline__ int refl2(int i, int n) {
  // jnp.pad mode='reflect' with pad 2: -1->1, -2->2, n->n-2, n+1->n-3
  if (i < 0) i = -i;
  if (i >= n) i = 2 * n - 2 - i;
  return i;
}

// Normalized 5-tap Gaussian (sigma=1, taps at -2..2): band value k[d] for d in [0,4], else 0.
// Branch-free: nested ternaries lower to v_cmp + v_cndmask (no EXEC manipulation).
__device__ __forceinline__ float bandw(int d) {
  float w = (d == 0 || d == 4) ? 0.05448868454964294f : 0.0f;
  w = (d == 1 || d == 3) ? 0.24420134200323332f : w;
  w = (d == 2) ? 0.40261994689424735f : w;
  return w;
}

// ---------------- blur kernels (f32 WMMA 16x16x4) ----------------
// One wave32 computes one 16x16 output tile of one channel.
// Vertical pass: D(16x16) = Wv(16x20) * X(20x16), Wv[m,j] = k[j-m]
// Horizontal  : D(16x16) = X(16x20) * Wh(20x16), Wh[j,n] = k[j-n]
// K=20 -> 5 chained V_WMMA_F32_16X16X4_F32.
// A 16x4 layout: lane L: m=L&15, VGPR0/1 hold K = 2*(L>>4) + {0,1}.
// B 4x16 layout: lane L: n=L&15, VGPR0/1 hold K = 2*(L>>4) + {0,1}.
// C/D layout: VGPR j -> (M = 8*(L>>4) + j, N = L&15).

__global__ __launch_bounds__(256) void fv_blur_v(const float* __restrict__ in,
                                                 float* __restrict__ out, int N) {
  const int lane = threadIdx.x & 31;
  const int wid  = threadIdx.x >> 5;
  const int TN = N >> 4;
  const int TC = TN * TN;
  const int total = 3 * TC;
  const int gw = blockIdx.x * 8 + wid;
  __shared__ float sw[8][20][16];

  const bool act = gw < total;
  int base = 0, r0 = 0, c0 = 0;
  if (act) {
    int ch = gw / TC;
    int t  = gw - ch * TC;
    int tr = t / TN, tc = t - tr * TN;
    r0 = tr << 4; c0 = tc << 4;
    base = ch * N * N;
    // load 20x16 window (rows r0-2 .. r0+17, reflect), coalesced
    for (int e = lane; e < 320; e += 32) {
      int wr = e >> 4, wc = e & 15;
      int gr = refl2(r0 - 2 + wr, N);
      sw[wid][wr][wc] = in[base + gr * N + c0 + wc];
    }
  }
  __syncthreads();
  if (!act) return;

  const int half = lane >> 4;      // selects K pair {0,1} vs {2,3} within each chunk
  const int lm   = lane & 15;      // A row m / B col n

  // Precompute all band weights (loop-invariant per lane) before the WMMA chain.
  float wa0[5], wa1[5];
#pragma unroll
  for (int c4 = 0; c4 < 5; ++c4) {
    int j0 = 4 * c4 + half * 2;
    wa0[c4] = bandw(j0 - lm);          // Wv[lm, j0]
    wa1[c4] = bandw(j0 + 1 - lm);      // Wv[lm, j0+1]
  }

  v8f acc = {};
#pragma unroll
  for (int c4 = 0; c4 < 5; ++c4) {
    int j0 = 4 * c4 + half * 2;
    v2f a, b;
    a.x = wa0[c4];
    a.y = wa1[c4];
    b.x = sw[wid][j0][lm];             // X[j0, lm]
    b.y = sw[wid][j0 + 1][lm];         // X[j0+1, lm]
    acc = __builtin_amdgcn_wmma_f32_16x16x4_f32(false, a, false, b, (short)0, acc, false, false);
  }
#pragma unroll
  for (int j = 0; j < 8; ++j)
    out[base + (r0 + half * 8 + j) * N + c0 + lm] = acc[j];
}

__global__ __launch_bounds__(256) void fv_blur_h(const float* __restrict__ in,
                                                 float* __restrict__ out, int N) {
  const int lane = threadIdx.x & 31;
  const int wid  = threadIdx.x >> 5;
  const int TN = N >> 4;
  const int TC = TN * TN;
  const int total = 3 * TC;
  const int gw = blockIdx.x * 8 + wid;
  __shared__ float sw[8][16][20];

  const bool act = gw < total;
  int base = 0, r0 = 0, c0 = 0;
  if (act) {
    int ch = gw / TC;
    int t  = gw - ch * TC;
    int tr = t / TN, tc = t - tr * TN;
    r0 = tr << 4; c0 = tc << 4;
    base = ch * N * N;
    // load 16x20 window (cols c0-2 .. c0+17, reflect)
    for (int e = lane; e < 320; e += 32) {
      int wr = e / 20, wc = e - wr * 20;
      int gc = refl2(c0 - 2 + wc, N);
      sw[wid][wr][wc] = in[base + (r0 + wr) * N + gc];
    }
  }
  __syncthreads();
  if (!act) return;

  const int half = lane >> 4;
  const int lm   = lane & 15;

  float wb0[5], wb1[5];
#pragma unroll
  for (int c4 = 0; c4 < 5; ++c4) {
    int j0 = 4 * c4 + half * 2;
    wb0[c4] = bandw(j0 - lm);          // Wh[j0, lm]
    wb1[c4] = bandw(j0 + 1 - lm);      // Wh[j0+1, lm]
  }

  v8f acc = {};
#pragma unroll
  for (int c4 = 0; c4 < 5; ++c4) {
    int j0 = 4 * c4 + half * 2;
    v2f a, b;
    a.x = sw[wid][lm][j0];             // X[lm, j0]
    a.y = sw[wid][lm][j0 + 1];         // X[lm, j0+1]
    b.x = wb0[c4];
    b.y = wb1[c4];
    acc = __builtin_amdgcn_wmma_f32_16x16x4_f32(false, a, false, b, (short)0, acc, false, false);
  }
#pragma unroll
  for (int j = 0; j < 8; ++j)
    out[base + (r0 + half * 8 + j) * N + c0 + lm] = acc[j];
}

// ---------------- align-corners bilinear resize (square, 3ch) ----------------

__global__ __launch_bounds__(256) void fv_resize_ac(const float* __restrict__ in,
                                                    float* __restrict__ out,
                                                    int Hi, int Ho, float ratio) {
  int idx = blockIdx.x * blockDim.x + threadIdx.x;
  int hw = Ho * Ho;
  if (idx >= 3 * hw) return;
  int ch  = idx / hw;
  int rem = idx - ch * hw;
  int oh  = rem / Ho;
  int ow  = rem - oh * Ho;

  float ph = ratio * (float)oh;
  int i0 = (int)floorf(ph);
  if (i0 < 0) i0 = 0; if (i0 > Hi - 1) i0 = Hi - 1;
  int i1 = i0 + 1; if (i1 > Hi - 1) i1 = Hi - 1;
  float wh = ph - (float)i0;

  float pw = ratio * (float)ow;
  int j0 = (int)floorf(pw);
  if (j0 < 0) j0 = 0; if (j0 > Hi - 1) j0 = Hi - 1;
  int j1 = j0 + 1; if (j1 > Hi - 1) j1 = Hi - 1;
  float ww = pw - (float)j0;

  int base = ch * Hi * Hi;
  float x00 = in[base + i0 * Hi + j0];
  float x10 = in[base + i1 * Hi + j0];
  float x01 = in[base + i0 * Hi + j1];
  float x11 = in[base + i1 * Hi + j1];
  float v0 = x00 * (1.0f - wh) + x10 * wh;   // H axis first (matches reference order)
  float v1 = x01 * (1.0f - wh) + x11 * wh;
  out[idx] = v0 * (1.0f - ww) + v1 * ww;
}

// ---------------- per-pixel blend accumulate ----------------
// Only Ms[L-1]=Bs[L-1] and Ms[L]=1-Bs[L-1] are nonzero at a pixel (L in 1..5).

__global__ __launch_bounds__(256) void fv_accum(const float* __restrict__ A,
                                                float* __restrict__ out,
                                                int level, int first) {
  const int HW = 2048 * 2048;
  int idx = blockIdx.x * blockDim.x + threadIdx.x;
  if (idx >= HW) return;
  int h = idx >> 11, w = idx & 2047;

  float dx = (float)h - 1024.0f, dy = (float)w - 1024.0f;
  float theta = sqrtf(dx * dx + dy * dy) / 7.5f;
  float R = 2.5f / (theta + 2.5f);

  const float W0  = 0.4806756f;        // sqrt(ln2/3)
  const float om1 = 2.0f * W0, om2 = W0, om3 = 0.5f * W0, om4 = 0.25f * W0;
  int L = 0;                           // sequential overwrite, later i wins (as in reference)
  if (R >= om1 && R <= 1.0f) L = 1;
  if (R >= om2 && R <= om1)  L = 2;
  if (R >= om3 && R <= om2)  L = 3;
  if (R >= om4 && R <= om3)  L = 4;
  if (R >= 0.f && R <= om4)  L = 5;

  // Ts[j] = exp(-(2^(j-2) * R)^2 * 3), Ts[5] = 0; need Ts[L-1], Ts[L]
  float sA = (L == 1) ? 0.25f : (L == 2) ? 0.5f : (L == 3) ? 1.0f : (L == 4) ? 2.0f : 4.0f;
  float uA = sA * R;
  float tA = expf(-((uA * uA) * 3.0f));
  float tB = 0.0f;
  if (L < 5) { float uB = (sA * 2.0f) * R; tB = expf(-((uB * uB) * 3.0f)); }
  float B = (0.5f - tB) / (tA - tB + 1e-5f);

  float M = 0.0f;
  if (level == L)     M = (level == 0) ? 1.0f : (1.0f - B);
  if (level == L - 1) M = B;           // L == level+1 (implies level < 5)

  if (first) {
#pragma unroll
    for (int c = 0; c < 3; ++c) out[c * HW + idx] = M * A[c * HW + idx];
  } else if (M != 0.0f) {
#pragma unroll
    for (int c = 0; c < 3; ++c) out[c * HW + idx] += M * A[c * HW + idx];
  }
}

// ---------------- host side ----------------

extern "C" void kernel_launch(void* const* d_in, const int* in_sizes, int n_in,
                              void* d_out, int out_size, void* d_ws, size_t ws_size,
                              hipStream_t stream) {
  (void)in_sizes; (void)n_in; (void)out_size; (void)ws_size;
  const float* img = (const float*)d_in[0];
  float* out = (float*)d_out;
  float* ws  = (float*)d_ws;

  const int H = 2048;
  float* bufA = ws;
  float* bufB = bufA + 3 * H * H;
  float* pyr[6]; pyr[0] = nullptr;
  {
    float* p = bufB + 3 * H * H;
    for (int i = 1; i <= 5; ++i) { int n = H >> i; pyr[i] = p; p += 3 * n * n; }
  }
  // total workspace used: 2*48MB + ~16.8MB ≈ 112.4 MB

  // ---- build Gaussian pyramid: blurH -> blurV -> 0.5x align-corners resize ----
  const float* cur = img;
  for (int i = 0; i < 5; ++i) {
    int N = H >> i;
    int waves  = 3 * (N / 16) * (N / 16);
    int blocks = (waves + 7) / 8;
    fv_blur_h<<<blocks, 256, 0, stream>>>(cur, bufA, N);
    fv_blur_v<<<blocks, 256, 0, stream>>>(bufA, bufB, N);
    int No = N / 2;
    int tot = 3 * No * No;
    float ratio = (float)((double)(N - 1) / (double)(No - 1));
    fv_resize_ac<<<(tot + 255) / 256, 256, 0, stream>>>(bufB, pyr[i + 1], N, No, ratio);
    cur = pyr[i + 1];
  }

  // ---- blend: out = sum_i Ms[i] * As[i], As[i] materialized via 2x doubling chain ----
  const int hw = H * H;
  fv_accum<<<(hw + 255) / 256, 256, 0, stream>>>(img, out, 0, 1);
  for (int i = 1; i <= 5; ++i) {
    const float* src = pyr[i];
    int n = H >> i;
    float* dsts[2] = {bufA, bufB};
    int pp = 0;
    while (n < H) {
      int no = n * 2;
      int tot = 3 * no * no;
      float ratio = (float)((double)(n - 1) / (double)(no - 1));
      fv_resize_ac<<<(tot + 255) / 256, 256, 0, stream>>>(src, dsts[pp], n, no, ratio);
      src = dsts[pp]; pp ^= 1; n = no;
    }
    // final _resize_ac(Gu, H, W) is identity (ratio==1, exact integer positions) -> skipped
    fv_accum<<<(hw + 255) / 256, 256, 0, stream>>>(src, out, i, 0);
  }
}